// TitansMemory_188978561365
// MI455X (gfx1250) — compile-verified
//
#include <hip/hip_runtime.h>
#include <stdint.h>

typedef __attribute__((ext_vector_type(2))) float v2f;
typedef __attribute__((ext_vector_type(4))) float v4f;
typedef __attribute__((ext_vector_type(8))) float v8f;

#define B_   16
#define L_   8192
#define D_   128     // DK == DV == 128
#define T_   16      // chunk length
#define NC_  (L_ / T_)

#define ALPHA_ 0.02f
#define LR_    0.1f
#define BETA_  0.9f

// ---- single LDS arena (float offsets); arena sits at LDS byte offset 0 ----
#define OFF_W     0        // 16x128 W rows
#define OFF_M     2048     // 16x128 momentum rows
#define OFF_K     4096     // 2 x 16x128 k chunk (double buffered)
#define OFF_Q     8192     // 2 x 16x128 q chunk (double buffered)
#define OFF_V     12288    // 2 x 16x16  v sub-tile (double buffered)
#define OFF_C     12800    // 16x16 rhs c[t][row]
#define OFF_R     13056    // 16x16 solved r[t][row]
#define OFF_L     13312    // 16x16 lower-tri couplings [u][t]
#define OFF_MT    13568    // 16x16 masked+scaled G [u][t]
#define OFF_PSI   13824    // psi[t][u]
#define OFF_PHI   14080
#define OFF_PHIM1 14096
#define OFF_POWA  14112    // a^t, t=0..16
#define OFF_PSIW  14129    // psi[15][u]
#define OFF_PSIB  14145    // beta^(15-u)
#define SMEM_TOT  14164

__device__ __forceinline__ v8f wmma4(v2f a, v2f b, v8f c) {
  // D = A(16x4 f32) * B(4x16 f32) + C(16x16 f32)
  return __builtin_amdgcn_wmma_f32_16x16x4_f32(false, a, false, b, (short)0, c,
                                               false, false);
}

__device__ __forceinline__ v8f zero8() {
  v8f z;
#pragma unroll
  for (int i = 0; i < 8; ++i) z[i] = 0.f;
  return z;
}

__device__ __forceinline__ float ipow(float x, int n) {
  float r = 1.f;
  for (int i = 0; i < n; ++i) r *= x;
  return r;
}

// gfx1250 async copy: global -> LDS, 16 bytes, tracked by ASYNCcnt.
__device__ __forceinline__ void async_b128(uint32_t lds_byte_off,
                                           const float* gptr) {
  asm volatile("global_load_async_to_lds_b128 %0, %1, off"
               :: "v"(lds_byte_off), "v"((uint64_t)(uintptr_t)gptr)
               : "memory");
}

// Issue one chunk's K/Q/V staging: 34 async ops per lane (in-order completion).
__device__ __forceinline__ void issue_chunk_async(const float* __restrict__ Kg,
                                                  const float* __restrict__ Qg,
                                                  const float* __restrict__ Vg,
                                                  long cbase, int buf, int lane,
                                                  int rowbase) {
  const uint32_t kbase = (uint32_t)(OFF_K + buf * 2048) * 4u;
  const uint32_t qbase = (uint32_t)(OFF_Q + buf * 2048) * 4u;
  const uint32_t vbase = (uint32_t)(OFF_V + buf * 256) * 4u;
#pragma unroll
  for (int i = 0; i < 16; ++i) {
    const int idx = i * 128 + lane * 4;
    async_b128(kbase + (uint32_t)idx * 4u, Kg + cbase + idx);
    async_b128(qbase + (uint32_t)idx * 4u, Qg + cbase + idx);
  }
#pragma unroll
  for (int i = 0; i < 2; ++i) {          // only the 16 dv columns we own
    const int f = lane * 8 + i * 4;      // 0..255
    const int t = f >> 4, c = f & 15;
    async_b128(vbase + (uint32_t)f * 4u, Vg + cbase + t * 128 + rowbase + c);
  }
}

// One wave (32 threads) per block; block owns 16 dv-rows of one batch.
// grid = B_ * (D_/16) = 128 blocks.
__launch_bounds__(32)
__global__ void titans_chunked_wmma(const float* __restrict__ Qg,
                                    const float* __restrict__ Kg,
                                    const float* __restrict__ Vg,
                                    float* __restrict__ Yg) {
  __shared__ __align__(16) float smem[SMEM_TOT];
  float* const sW   = smem + OFF_W;
  float* const sMst = smem + OFF_M;

  const int lane    = threadIdx.x;
  const int batch   = blockIdx.x >> 3;
  const int rowbase = (blockIdx.x & 7) * 16;
  const int m    = lane & 15;   // WMMA M/N index within a half
  const int h    = lane >> 4;   // lane half
  const int koff = 2 * h;       // A/B K offset for this half

  const float aC  = 1.f - ALPHA_;        // W decay per step
  const float g   = 2.f * (1.f - BETA_); // grad->momentum gain
  const float lrg = LR_ * g;

  const long baseIn = (long)batch * L_ * D_;

  // kick off chunk 0 staging immediately; overlap with init below
  issue_chunk_async(Kg, Qg, Vg, baseIn, 0, lane, rowbase);

  // ---- init state + scalar tables ----
  for (int i = lane; i < 16 * D_; i += 32) { sW[i] = 0.f; sMst[i] = 0.f; }
  if (lane < 16) {
    const int t = lane;
    float phi = 0.f;
    for (int s = 0; s <= t; ++s) phi += ipow(aC, t - s) * ipow(BETA_, s + 1);
    smem[OFF_PHI + t] = phi;
    float phim = 0.f;
    for (int s = 0; s < t; ++s) phim += ipow(aC, t - 1 - s) * ipow(BETA_, s + 1);
    smem[OFF_PHIM1 + t] = phim;
    for (int u = 0; u < 16; ++u) {
      float ps = 0.f;
      if (u <= t)
        for (int s = u; s <= t; ++s) ps += ipow(aC, t - s) * ipow(BETA_, s - u);
      smem[OFF_PSI + t * 16 + u] = ps;
    }
    const int u = lane;
    float pw = 0.f;
    for (int s = u; s <= 15; ++s) pw += ipow(aC, 15 - s) * ipow(BETA_, s - u);
    smem[OFF_PSIW + u] = pw;
    smem[OFF_PSIB + u] = ipow(BETA_, 15 - u);
  }
  if (lane < 17) smem[OFF_POWA + lane] = ipow(aC, lane);
  __syncthreads();

  const float powa16 = smem[OFF_POWA + 16];
  const float powb16 = ipow(BETA_, 16);
  const float phi15  = smem[OFF_PHI + 15];

  for (int ch = 0; ch < NC_; ++ch) {
    const long cbase = baseIn + (long)ch * T_ * D_;
    const float* sKc = smem + OFF_K + (ch & 1) * 2048;
    const float* sQc = smem + OFF_Q + (ch & 1) * 2048;
    const float* sVc = smem + OFF_V + (ch & 1) * 256;

    // double buffer: start filling the other buffer, then fence this one.
    if (ch + 1 < NC_) {
      issue_chunk_async(Kg, Qg, Vg, cbase + T_ * D_, (ch + 1) & 1, lane, rowbase);
      asm volatile("s_wait_asynccnt 34" ::: "memory");
    } else {
      asm volatile("s_wait_asynccnt 0" ::: "memory");
    }
    __syncthreads();

    // ---- fused chunk matmuls over K-dim (f32 WMMA 16x16x4):
    //   U=W0*K^T  P=m0*K^T  U'=W0*Q^T  P'=m0*Q^T  A=K*K^T  G=K*Q^T
    v8f U = zero8(), P = zero8(), Up = zero8(), Pp = zero8();
    v8f Ac = zero8(), Gc = zero8();
#pragma unroll
    for (int kk = 0; kk < 32; ++kk) {
      const int d = kk * 4 + koff;
      const v2f bK = *(const v2f*)&sKc[m * D_ + d];  // K as A-op == K^T as B-op
      const v2f bQ = *(const v2f*)&sQc[m * D_ + d];
      const v2f aW = *(const v2f*)&sW[m * D_ + d];
      const v2f aM = *(const v2f*)&sMst[m * D_ + d];
      U  = wmma4(aW, bK, U);
      P  = wmma4(aM, bK, P);
      Up = wmma4(aW, bQ, Up);
      Pp = wmma4(aM, bQ, Pp);
      Ac = wmma4(bK, bK, Ac);
      Gc = wmma4(bK, bQ, Gc);
    }

    // ---- rhs c_t and coupling tables (C layout: M = r+8h, N = m) ----
#pragma unroll
    for (int r = 0; r < 8; ++r) {
      const int row = r + 8 * h;  // local row (or u index for Ac/Gc)
      const int t   = m;
      smem[OFF_C + t * 16 + row] =
          smem[OFF_POWA + t] * U[r] - LR_ * smem[OFF_PHIM1 + t] * P[r]
          - sVc[t * 16 + row];
      const int u = row;
      smem[OFF_L + u * 16 + t] =
          (u < t) ? (lrg * smem[OFF_PSI + (t - 1) * 16 + u] * Ac[r]) : 0.f;
      smem[OFF_MT + u * 16 + t] =
          (u <= t) ? (smem[OFF_PSI + t * 16 + u] * Gc[r]) : 0.f;
    }
    __syncthreads();

    // ---- 16-step forward substitution: r_t = c_t - sum_{u<t} L[u][t] r_u ----
    if (lane < 16) {
      const int row = lane;
      for (int t = 0; t < 16; ++t) {
        float rr = smem[OFF_C + t * 16 + row];
        for (int u = 0; u < t; ++u)
          rr -= smem[OFF_L + u * 16 + t] * smem[OFF_R + u * 16 + row];
        smem[OFF_R + t * 16 + row] = rr;
      }
    }
    __syncthreads();

    // ---- outputs: Ym = R * (masked psi.G); y = a^{t+1}U' - lr phi_t P' - lrg Ym
    v8f Ym = zero8();
#pragma unroll
    for (int kk = 0; kk < 4; ++kk) {
      const int u = kk * 4 + koff;
      v2f aR; aR[0] = smem[OFF_R + u * 16 + m];
      aR[1]         = smem[OFF_R + (u + 1) * 16 + m];
      v2f bM; bM[0] = smem[OFF_MT + u * 16 + m];
      bM[1]         = smem[OFF_MT + (u + 1) * 16 + m];
      Ym = wmma4(aR, bM, Ym);
    }
#pragma unroll
    for (int r = 0; r < 8; ++r) {
      const int row = r + 8 * h;
      const int t   = m;
      const float y = smem[OFF_POWA + t + 1] * Up[r]
                    - LR_ * smem[OFF_PHI + t] * Pp[r] - lrg * Ym[r];
      Yg[cbase + (long)t * D_ + rowbase + row] = y;
    }

    // ---- state update: W' = a^16 W - lr phi15 m - lrg R diag(psiW) K
    //                    m' = b^16 m + g  R diag(psiB) K
#pragma unroll
    for (int kt = 0; kt < 8; ++kt) {
      v8f CW = zero8(), CM = zero8();
#pragma unroll
      for (int kk = 0; kk < 4; ++kk) {
        const int u = kk * 4 + koff;
        const float r0 = smem[OFF_R + u * 16 + m];
        const float r1 = smem[OFF_R + (u + 1) * 16 + m];
        v2f aRW; aRW[0] = r0 * smem[OFF_PSIW + u];
        aRW[1]          = r1 * smem[OFF_PSIW + u + 1];
        v2f aRM; aRM[0] = r0 * smem[OFF_PSIB + u];
        aRM[1]          = r1 * smem[OFF_PSIB + u + 1];
        v2f bK2; bK2[0] = sKc[u * D_ + kt * 16 + m];
        bK2[1]          = sKc[(u + 1) * D_ + kt * 16 + m];
        CW = wmma4(aRW, bK2, CW);
        CM = wmma4(aRM, bK2, CM);
      }
#pragma unroll
      for (int r = 0; r < 8; ++r) {
        const int row = r + 8 * h;
        const int idx = row * D_ + kt * 16 + m;
        const float w0  = sW[idx];
        const float m0v = sMst[idx];
        sMst[idx] = powb16 * m0v + g * CM[r];
        sW[idx]   = powa16 * w0 - LR_ * phi15 * m0v - lrg * CW[r];
      }
    }
  }
}

extern "C" void kernel_launch(void* const* d_in, const int* in_sizes, int n_in,
                              void* d_out, int out_size, void* d_ws, size_t ws_size,
                              hipStream_t stream) {
  (void)in_sizes; (void)n_in; (void)out_size; (void)d_ws; (void)ws_size;
  const float* Q = (const float*)d_in[0];
  const float* K = (const float*)d_in[1];
  const float* V = (const float*)d_in[2];
  float* Y = (float*)d_out;
  dim3 grid(B_ * (D_ / 16));   // 128 blocks: 16 batches x 8 row-groups
  dim3 block(32);              // one wave32 per block
  titans_chunked_wmma<<<grid, block, 0, stream>>>(Q, K, V, Y);
}